// Sampler_86990267613915
// MI455X (gfx1250) — compile-verified
//
#include <hip/hip_runtime.h>
#include <math.h>
#include <stdint.h>

#define NT 256              // 8 waves of 32 on CDNA5
#define CHUNK 2048          // 8KB staging chunk (double buffered)
#define NBINS 4096          // 12-bit monotone-key histogram
#define SEG (NBINS / NT)    // bins per thread in the crossing scan

typedef unsigned int u32x4 __attribute__((ext_vector_type(4)));
typedef int i32x8 __attribute__((ext_vector_type(8)));
typedef int i32x4 __attribute__((ext_vector_type(4)));

// Monotone float -> uint key: order-preserving for all finite floats and +/-inf.
__device__ __forceinline__ unsigned fkey(float x) {
  unsigned u = __float_as_uint(x);
  return (u & 0x80000000u) ? ~u : (u | 0x80000000u);
}

// ---- Tensor Data Mover: DMA a contiguous run of `nelem` f32 into LDS ----
// D# per CDNA5 ISA ch.8: group0 {count=1, lds_addr, global_addr, type=2},
// group1 {data_size=2 (4B), tensor_dim0=nelem, tensor_dim1=1, tile_dim0=nelem,
//         tensor_dim0_stride=nelem}; groups 2/3 zero (<=2D tensor).
// This toolchain's builtin takes 6 args (extra int32x8 group, then cpol).
__device__ __forceinline__ void tdm_load_row_chunk(const float* gptr,
                                                   unsigned lds_off, int nelem) {
  unsigned long long ga = (unsigned long long)(size_t)gptr;
  unsigned n = (unsigned)nelem;
  u32x4 g0;
  g0.x = 0x1u;                                   // count=1, is_restore=0, no gather
  g0.y = lds_off;                                // LDS byte address
  g0.z = (unsigned)(ga & 0xFFFFFFFFull);         // global_addr[31:0]
  g0.w = (unsigned)((ga >> 32) & 0x01FFFFFFull) | (0x2u << 30); // addr[56:32]|type=2
  i32x8 g1;
  g1[0] = (int)(2u << 16);                       // wg_mask=0, data_size=2 (4 bytes)
  g1[1] = (int)((n & 0xFFFFu) << 16);            // tensor_dim0[15:0]
  g1[2] = (int)(((n >> 16) & 0xFFFFu) | (1u << 16)); // tensor_dim0[31:16], tensor_dim1=1
  g1[3] = (int)((n & 0xFFFFu) << 16);            // tile_dim0 = nelem (<=65535)
  g1[4] = 0;                                     // tile_dim1=0, tile_dim2=0 (unused)
  g1[5] = (int)n;                                // tensor_dim0_stride[31:0]
  g1[6] = 0;
  g1[7] = 0;
  i32x4 z4 = {0, 0, 0, 0};
  i32x8 z8 = {0, 0, 0, 0, 0, 0, 0, 0};
  __builtin_amdgcn_tensor_load_to_lds(g0, g1, z4, z4, z8, 0);
}

// ---- Stream one row through double-buffered LDS via the TDM. ----
// wave 0 issues DMA for chunk c+1, waits TENSORcnt<=1 (in-order per wave =>
// chunk c has landed), then all 8 waves consume chunk c after the barrier.
template <typename F>
__device__ __forceinline__ void stream_row(const float* rowBase, int V,
                                           float* buf0, float* buf1,
                                           int tid, bool w0, F&& consume) {
  int nch = (V + CHUNK - 1) / CHUNK;
  if (w0) tdm_load_row_chunk(rowBase, (unsigned)(size_t)buf0, min(CHUNK, V));
  for (int c = 0; c < nch; ++c) {
    int off = c * CHUNK;
    int len = min(CHUNK, V - off);
    float* cur = (c & 1) ? buf1 : buf0;
    if (w0) {
      if (c + 1 < nch) {
        int off2 = (c + 1) * CHUNK;
        float* nxt = ((c + 1) & 1) ? buf1 : buf0;
        tdm_load_row_chunk(rowBase + off2, (unsigned)(size_t)nxt,
                           min(CHUNK, V - off2));
        __builtin_amdgcn_s_wait_tensorcnt(1);    // chunk c complete (in-order)
      } else {
        __builtin_amdgcn_s_wait_tensorcnt(0);
      }
    }
    __syncthreads();
    consume(cur, off, len);
    __syncthreads();                             // done with buf before re-fill
  }
}

// ---- Find the bin where the DESCENDING cumulative of s_hist first reaches
// `target`. Returns bin index (or -1) and residual target inside that bin. ----
__device__ void cross_desc(const float* s_hist, float target, int tid,
                           float* s_part, int* s_bin, float* s_resid) {
  float part = 0.f;
  int rbase = tid * SEG;                         // r = 0 is the HIGHEST bin
  for (int j = 0; j < SEG; ++j) part += s_hist[NBINS - 1 - (rbase + j)];
  s_part[tid] = part;
  if (tid == 0) *s_bin = -1;
  __syncthreads();
  for (int o = 1; o < NT; o <<= 1) {             // Hillis-Steele inclusive scan
    float add = (tid >= o) ? s_part[tid - o] : 0.f;
    __syncthreads();
    s_part[tid] += add;
    __syncthreads();
  }
  float incl = s_part[tid];
  float pre = incl - part;                       // exclusive prefix (higher bins)
  if (pre < target && incl >= target) {          // unique qualifying thread
    float cum = pre;
    for (int j = 0; j < SEG; ++j) {
      int bin = NBINS - 1 - (rbase + j);
      float e = s_hist[bin];
      if (cum + e >= target || j == SEG - 1) {
        *s_bin = bin;
        *s_resid = target - cum;
        break;
      }
      cum += e;
    }
  }
  __syncthreads();
}

// ---- Kernel 1: per-row stats -> {rowmax, final keep-key, greedy argmax} ----
__global__ void __launch_bounds__(NT)
sampler_stats_kernel(const float* __restrict__ logits,
                     const float* __restrict__ temps,
                     const float* __restrict__ top_ps,
                     const int* __restrict__ top_ks,
                     const float* __restrict__ min_ps,
                     float* ws_rowmax, unsigned* ws_key, int* ws_greedy, int V) {
  __shared__ float buf0[CHUNK], buf1[CHUNK];
  __shared__ float s_hist[NBINS];
  __shared__ float s_part[NT];
  __shared__ float s_rv[NT];
  __shared__ int s_ri[NT];
  __shared__ int s_bin;
  __shared__ float s_resid;
  __shared__ float s_acc;

  const int row = blockIdx.x, tid = threadIdx.x;
  const bool w0 = (tid < 32);
  const float* rb = logits + (size_t)row * V;
  const float invT = 1.0f / temps[row];

  for (int i = tid; i < NBINS; i += NT) s_hist[i] = 0.f;
  __syncthreads();

  // Pass 1: 12-bit count histogram + rowmax/argmax (greedy).
  float mv = -__builtin_inff();
  int mi = 0;
  stream_row(rb, V, buf0, buf1, tid, w0, [&](float* cur, int off, int len) {
    for (int j = tid; j < len; j += NT) {
      float x = cur[j] * invT;
      atomicAdd(&s_hist[fkey(x) >> 20], 1.0f);
      int gi = off + j;
      if (x > mv || (x == mv && gi < mi)) { mv = x; mi = gi; }
    }
  });
  s_rv[tid] = mv; s_ri[tid] = mi;
  __syncthreads();
  for (int o = NT / 2; o > 0; o >>= 1) {
    if (tid < o) {
      float v2 = s_rv[tid + o]; int i2 = s_ri[tid + o];
      if (v2 > s_rv[tid] || (v2 == s_rv[tid] && i2 < s_ri[tid])) {
        s_rv[tid] = v2; s_ri[tid] = i2;
      }
    }
    __syncthreads();
  }
  const float rowmax = s_rv[0];
  const int greedy = s_ri[0];
  __syncthreads();

  // top-k: bin of the k-th largest, then 12-bit refinement => 24-bit key thr.
  float kf = (float)max(1, top_ks[row]);
  cross_desc(s_hist, kf, tid, s_part, &s_bin, &s_resid);
  int bk = s_bin; float rk = s_resid;
  unsigned key_tk = 0;
  if (bk >= 0) {
    for (int i = tid; i < NBINS; i += NT) s_hist[i] = 0.f;
    __syncthreads();
    stream_row(rb, V, buf0, buf1, tid, w0, [&](float* cur, int off, int len) {
      (void)off;
      for (int j = tid; j < len; j += NT) {
        unsigned k = fkey(cur[j] * invT);
        if ((int)(k >> 20) == bk) atomicAdd(&s_hist[(k >> 8) & 0xFFFu], 1.0f);
      }
    });
    cross_desc(s_hist, rk, tid, s_part, &s_bin, &s_resid);
    int sk = s_bin;
    key_tk = ((unsigned)bk << 20) | ((sk >= 0 ? (unsigned)sk : 0u) << 8);
  }

  // top-p: mass histogram over the top-k-kept set, then refinement.
  for (int i = tid; i < NBINS; i += NT) s_hist[i] = 0.f;
  if (tid == 0) s_acc = 0.f;
  __syncthreads();
  stream_row(rb, V, buf0, buf1, tid, w0, [&](float* cur, int off, int len) {
    (void)off;
    float lz = 0.f;
    for (int j = tid; j < len; j += NT) {
      float x = cur[j] * invT;
      unsigned k = fkey(x);
      if (k >= key_tk) {
        float m = __expf(x - rowmax);
        atomicAdd(&s_hist[k >> 20], m);
        lz += m;
      }
    }
    atomicAdd(&s_acc, lz);
  });
  float targ = top_ps[row] * s_acc;              // p * Z  (Z over top-k kept)
  cross_desc(s_hist, targ, tid, s_part, &s_bin, &s_resid);
  int bp = s_bin; float rp = s_resid;
  unsigned key_tp = 0;
  if (bp >= 0) {
    for (int i = tid; i < NBINS; i += NT) s_hist[i] = 0.f;
    __syncthreads();
    stream_row(rb, V, buf0, buf1, tid, w0, [&](float* cur, int off, int len) {
      (void)off;
      for (int j = tid; j < len; j += NT) {
        float x = cur[j] * invT;
        unsigned k = fkey(x);
        if (k >= key_tk && (int)(k >> 20) == bp)
          atomicAdd(&s_hist[(k >> 8) & 0xFFFu], __expf(x - rowmax));
      }
    });
    cross_desc(s_hist, rp, tid, s_part, &s_bin, &s_resid);
    int sp = s_bin;
    key_tp = ((unsigned)bp << 20) | ((sp >= 0 ? (unsigned)sp : 0u) << 8);
  }

  // min-p is closed-form: keep iff x >= rowmax + ln(min_p).
  float mp = min_ps[row];
  unsigned key_mp = (mp > 0.f) ? fkey(rowmax + __logf(mp)) : 0u;

  unsigned keyF = max(key_tk, max(key_tp, key_mp));
  if (tid == 0) {
    ws_rowmax[row] = rowmax;
    ws_key[row] = keyF;
    ws_greedy[row] = greedy;
  }
}

// ---- Kernel 2: exact Z over kept set, multinomial argmax, logprobs out ----
__global__ void __launch_bounds__(NT)
sampler_final_kernel(const float* __restrict__ logits,
                     const float* __restrict__ temps,
                     const float* __restrict__ expn,
                     const float* __restrict__ ws_rowmax,
                     const unsigned* __restrict__ ws_key,
                     const int* __restrict__ ws_greedy,
                     float* __restrict__ out, int B, int V) {
  __shared__ float buf0[CHUNK], buf1[CHUNK];
  __shared__ float s_rv[NT];
  __shared__ int s_ri[NT];
  __shared__ float s_acc;

  const int row = blockIdx.x, tid = threadIdx.x;
  const bool w0 = (tid < 32);
  const float* rb = logits + (size_t)row * V;
  const float* en = expn + (size_t)row * V;
  const float invT = 1.0f / temps[row];
  const float rowmax = ws_rowmax[row];
  const unsigned keyF = ws_key[row];

  // Pass A: Z over the kept set + multinomial argmax of x - ln(q) (Gumbel trick).
  float Zl = 0.f, bs = -__builtin_inff();
  int bi = 0;
  stream_row(rb, V, buf0, buf1, tid, w0, [&](float* cur, int off, int len) {
    __builtin_prefetch(en + off + len, 0, 1);    // global_prefetch_b8 for noise
    for (int j = tid; j < len; j += NT) {
      float x = cur[j] * invT;
      if (fkey(x) >= keyF) {
        Zl += __expf(x - rowmax);
        int gi = off + j;
        float sc = x - __logf(en[gi]);
        if (sc > bs || (sc == bs && gi < bi)) { bs = sc; bi = gi; }
      }
    }
  });
  if (tid == 0) s_acc = 0.f;
  __syncthreads();
  atomicAdd(&s_acc, Zl);
  s_rv[tid] = bs; s_ri[tid] = bi;
  __syncthreads();
  const float lnZ = __logf(s_acc);
  for (int o = NT / 2; o > 0; o >>= 1) {
    if (tid < o) {
      float v2 = s_rv[tid + o]; int i2 = s_ri[tid + o];
      if (v2 > s_rv[tid] || (v2 == s_rv[tid] && i2 < s_ri[tid])) {
        s_rv[tid] = v2; s_ri[tid] = i2;
      }
    }
    __syncthreads();
  }
  if (tid == 0) {
    out[(size_t)B * V + row] = (float)s_ri[0];            // multinomial [B,1]
    out[(size_t)B * V + B + row] = (float)ws_greedy[row]; // greedy [B]
  }
  __syncthreads();

  // Pass B: logprobs = x - rowmax - lnZ on kept, -inf elsewhere.
  float* orow = out + (size_t)row * V;
  stream_row(rb, V, buf0, buf1, tid, w0, [&](float* cur, int off, int len) {
    for (int j = tid; j < len; j += NT) {
      float x = cur[j] * invT;
      orow[off + j] = (fkey(x) >= keyF) ? (x - rowmax - lnZ) : -__builtin_inff();
    }
  });
}

extern "C" void kernel_launch(void* const* d_in, const int* in_sizes, int n_in,
                              void* d_out, int out_size, void* d_ws, size_t ws_size,
                              hipStream_t stream) {
  (void)n_in; (void)out_size; (void)ws_size;
  const int B = in_sizes[1];
  const int V = in_sizes[0] / B;
  const float* logits = (const float*)d_in[0];
  const float* temps  = (const float*)d_in[1];
  const float* topps  = (const float*)d_in[2];
  const int*   topks  = (const int*)d_in[3];
  const float* minps  = (const float*)d_in[4];
  const float* expn   = (const float*)d_in[5];

  float*    ws_rowmax = (float*)d_ws;
  unsigned* ws_key    = (unsigned*)((char*)d_ws + (size_t)B * 4);
  int*      ws_greedy = (int*)((char*)d_ws + (size_t)B * 8);

  sampler_stats_kernel<<<B, NT, 0, stream>>>(logits, temps, topps, topks, minps,
                                             ws_rowmax, ws_key, ws_greedy, V);
  sampler_final_kernel<<<B, NT, 0, stream>>>(logits, temps, expn, ws_rowmax,
                                             ws_key, ws_greedy, (float*)d_out,
                                             B, V);
}